// QuantConv_PW_54107997995723
// MI455X (gfx1250) — compile-verified
//
#include <hip/hip_runtime.h>

// Problem constants (from reference)
#define BB      8
#define IN_P    16
#define OUT_P   32
#define REPC    16
#define PRODDW  256          // IN_P * REPC
#define HH      32
#define WW      32
#define HW      1024         // HH*WW

typedef __attribute__((ext_vector_type(2))) float v2f;
typedef __attribute__((ext_vector_type(8))) float v8f;

// ---------------------------------------------------------------------------
// Kernel 1: gated broadcast-multiply producing out1 (256 MB, streaming bound).
// One thread owns (b, i, hw): 16 x-loads -> registers, one 16-wide sum,
// then 32 (o) x 16 (r) gated multiplies, emitted as non-temporal stores
// (output exceeds the 192MB L2 and is never re-read).
// ---------------------------------------------------------------------------
__global__ __launch_bounds__(256) void out1_gate_kernel(
    const float* __restrict__ x,      // (B, 256, H, W)
    const float* __restrict__ w2,     // (32, 16) row-major (o, i)
    float* __restrict__ out1)         // (B, 8192, H, W)
{
    const int pos = blockIdx.x * 256 + threadIdx.x;  // 0..1023 spatial
    const int i   = blockIdx.y;                      // 0..15
    const int b   = blockIdx.z;                      // 0..7

    const float* xp = x + ((size_t)b * PRODDW + (size_t)i * REPC) * HW + pos;

    float xv[REPC];
    float s = 0.f;
#pragma unroll
    for (int r = 0; r < REPC; ++r) {
        xv[r] = __builtin_nontemporal_load(xp + (size_t)r * HW);  // read-once data
        s += xv[r];
    }

    float* op = out1 + ((size_t)b * (OUT_P * PRODDW) + (size_t)i * REPC) * HW + pos;

#pragma unroll 4
    for (int o = 0; o < OUT_P; ++o) {
        const float w   = w2[o * IN_P + i];        // uniform -> s_load path
        const float wg  = (w * s > 0.f) ? w : 0.f; // gate folded into the weight
        float* opo = op + (size_t)o * (PRODDW * HW);
#pragma unroll
        for (int r = 0; r < REPC; ++r) {
            __builtin_nontemporal_store(xv[r] * wg, opo + (size_t)r * HW);
        }
    }
}

// ---------------------------------------------------------------------------
// Kernel 2: dense 1x1 conv + ReLU via V_WMMA_F32_16X16X4_F32.
// GEMM: D(32 x 8192) = w2(32 x 16) * x2(16 x 8192), K tiled 4x4.
// One wave -> 32(o) x 16(pixel) tile: two v8f accumulators, 8 WMMAs.
//
// VGPR layouts per CDNA5 ISA 7.12.2:
//   A 16x4 f32 : lanes 0-15 & 16-31 both M = lane%16;
//                VGPR0 holds K = 0(lo half)/2(hi half), VGPR1 holds K = 1/3.
//   B 4x16 f32 : mirrored (N = lane%16, VGPR0 -> K=0/2, VGPR1 -> K=1/3).
//   C/D 16x16  : VGPR v, lanes 0-15: M=v, N=lane; lanes 16-31: M=v+8, N=lane-16.
// ---------------------------------------------------------------------------
__global__ __launch_bounds__(256) void out2_wmma_kernel(
    const float* __restrict__ x2,     // (B, 16, H, W)
    const float* __restrict__ w2,     // (32, 16)
    float* __restrict__ out2)         // (B, 32, H, W)
{
    const int lane  = threadIdx.x & 31;
    const int wave  = threadIdx.x >> 5;
    const int ptile = blockIdx.x * 8 + wave;   // 16-pixel tile id, 0..511
    const int pixb  = ptile * 16;              // HW%16==0 -> never crosses b
    const int b     = pixb / HW;
    const int hw    = pixb % HW;

    const int lmod  = lane & 15;
    const int khalf = (lane >> 4) << 1;        // 0 (lanes 0-15) or 2 (lanes 16-31)

    v8f acc0 = {};   // o = 0..15
    v8f acc1 = {};   // o = 16..31

    const float* xb = x2 + (size_t)b * IN_P * HW + hw + lmod;

#pragma unroll
    for (int k = 0; k < 4; ++k) {
        const int kb = k * 4;
        v2f a0, a1, bm;
        // A tiles (weights), M = o, K = i
        a0.x = w2[lmod * IN_P        + kb + khalf];
        a0.y = w2[lmod * IN_P        + kb + khalf + 1];
        a1.x = w2[(16 + lmod) * IN_P + kb + khalf];
        a1.y = w2[(16 + lmod) * IN_P + kb + khalf + 1];
        // B tile (activations), K = i, N = pixel
        bm.x = xb[(size_t)(kb + khalf)     * HW];
        bm.y = xb[(size_t)(kb + khalf + 1) * HW];

        acc0 = __builtin_amdgcn_wmma_f32_16x16x4_f32(false, a0, false, bm,
                                                     (short)0, acc0, false, false);
        acc1 = __builtin_amdgcn_wmma_f32_16x16x4_f32(false, a1, false, bm,
                                                     (short)0, acc1, false, false);
    }

    // Fused ReLU + store. M offset 0/8 selected by lane half.
    const int mofs = (lane >> 4) * 8;
    float* op = out2 + (size_t)b * OUT_P * HW + hw + lmod;
#pragma unroll
    for (int v = 0; v < 8; ++v) {
        float d0 = acc0[v]; d0 = d0 > 0.f ? d0 : 0.f;
        float d1 = acc1[v]; d1 = d1 > 0.f ? d1 : 0.f;
        op[(size_t)(mofs + v)      * HW] = d0;
        op[(size_t)(16 + mofs + v) * HW] = d1;
    }
}

extern "C" void kernel_launch(void* const* d_in, const int* in_sizes, int n_in,
                              void* d_out, int out_size, void* d_ws, size_t ws_size,
                              hipStream_t stream) {
    const float* x  = (const float*)d_in[0];   // (8, 256, 32, 32)
    const float* x2 = (const float*)d_in[1];   // (8, 16, 32, 32)
    const float* w2 = (const float*)d_in[2];   // (512,) == (32, 16)

    float* out1 = (float*)d_out;
    float* out2 = (float*)d_out + (size_t)BB * OUT_P * PRODDW * HW;  // +64Mi floats

    // out1: 4 x 16 x 8 = 512 blocks x 256 threads, each thread one (b,i,hw)
    dim3 g1(HW / 256, IN_P, BB);
    out1_gate_kernel<<<g1, 256, 0, stream>>>(x, w2, out1);

    // out2: 512 pixel tiles / 8 waves per block = 64 blocks
    out2_wmma_kernel<<<(BB * HW) / 128, 256, 0, stream>>>(x2, w2, out2);
}